// EGNN_Layer_51788715655765
// MI455X (gfx1250) — compile-verified
//
#include <hip/hip_runtime.h>
#include <hip/hip_bf16.h>

#define B_   8
#define N_   1024
#define M_   32
#define D_   128
#define E_   32
#define DD_  32
#define MSG_ 128
#define P_   128
#define KREST 192            // (2*D + DD + E) - D : neighbor-specific pair columns

typedef float v2f __attribute__((ext_vector_type(2)));
typedef float v8f __attribute__((ext_vector_type(8)));

// -ln(10000)/15 for the sinusoidal frequencies (half = 16)
#define NEG_LOG_THETA_OVER_HM1 (-0.6140226914650789f)

// 32x128 = A(32xK=128, row stride 128 in LDS) @ W(128x128 global, row major) + bias
// Each wave owns a 16-wide column slice (col0) and both 16-row tiles.
__device__ __forceinline__ void gemm32x128(const float* __restrict__ A,
                                           const float* __restrict__ W,
                                           const float* __restrict__ bias,
                                           float* __restrict__ Out,
                                           bool do_relu,
                                           int half, int nidx, int col0)
{
    v8f c0 = {}; v8f c1 = {};
    for (int k0 = 0; k0 < MSG_; k0 += 4) {
        const int kk = k0 + 2 * half;
        v2f a0; a0.x = A[nidx * MSG_ + kk];        a0.y = A[nidx * MSG_ + kk + 1];
        v2f a1; a1.x = A[(16 + nidx) * MSG_ + kk]; a1.y = A[(16 + nidx) * MSG_ + kk + 1];
        v2f bf; bf.x = W[kk * MSG_ + col0 + nidx];
                bf.y = W[(kk + 1) * MSG_ + col0 + nidx];
        c0 = __builtin_amdgcn_wmma_f32_16x16x4_f32(false, a0, false, bf, (short)0, c0, false, false);
        c1 = __builtin_amdgcn_wmma_f32_16x16x4_f32(false, a1, false, bf, (short)0, c1, false, false);
    }
    const float bv = bias[col0 + nidx];
#pragma unroll
    for (int v = 0; v < 8; ++v) {
        const int r = v + 8 * half;
        float o0 = c0[v] + bv;
        float o1 = c1[v] + bv;
        if (do_relu) { o0 = fmaxf(o0, 0.0f); o1 = fmaxf(o1, 0.0f); }
        Out[r * MSG_ + col0 + nidx]        = o0;
        Out[(16 + r) * MSG_ + col0 + nidx] = o1;
    }
}

__global__ __launch_bounds__(256)
void egnn_node_kernel(const float* __restrict__ emb,    const float* __restrict__ coords,
                      const float* __restrict__ pmask,  const float* __restrict__ edges,
                      const float* __restrict__ we_w1,  const float* __restrict__ we_b1,
                      const float* __restrict__ we_w2,  const float* __restrict__ we_b2,
                      const float* __restrict__ wx_w1,  const float* __restrict__ wx_b1,
                      const float* __restrict__ wx_w2,  const float* __restrict__ wx_b2,
                      const float* __restrict__ wh_w1,  const float* __restrict__ wh_b1,
                      const float* __restrict__ wh_w2,  const float* __restrict__ wh_b2,
                      const int*   __restrict__ nbr,
                      float* __restrict__ out_emb, float* __restrict__ out_coords)
{
    __shared__ float sXr[M_][KREST];   // [emb_j | sinus dists | edges]  (24 KB, reused as Hx)
    __shared__ float sH [M_][MSG_];    // relu(X @ we_w1 + b1)           (16 KB)
    __shared__ float sM [M_][MSG_];    // m_ij                           (16 KB)
    __shared__ float sEmb[D_];
    __shared__ float sMi [MSG_];
    __shared__ float sT1 [P_];
    __shared__ float sUnit[M_][3];
    __shared__ float sDist[M_];
    __shared__ float sPx [M_];
    __shared__ int   sNid[M_];

    const int p    = blockIdx.x;            // node index in [0, B*N)
    const int bIdx = p / N_;
    const int tid  = threadIdx.x;
    const int lane = tid & 31;
    const int wave = tid >> 5;
    const int half = lane >> 4;
    const int nidx = lane & 15;
    const int col0 = wave * 16;

    const float* embp = emb    + (size_t)p * D_;
    const float* cdp  = coords + (size_t)p * 3;
    const float* edp  = edges  + (size_t)p * (M_ * E_);
    const int*   nbp  = nbr    + (size_t)p * M_;

    // ---- Phase 1: cache emb_i, per-neighbor geometry --------------------
    if (tid < D_) sEmb[tid] = embp[tid];
    if (tid < M_) {
        const int g = bIdx * N_ + nbp[tid];          // global row of neighbor
        sNid[tid] = g;
        const float rx = cdp[0] - coords[(size_t)g * 3 + 0];
        const float ry = cdp[1] - coords[(size_t)g * 3 + 1];
        const float rz = cdp[2] - coords[(size_t)g * 3 + 2];
        const float d  = sqrtf(rx * rx + ry * ry + rz * rz);
        sDist[tid] = d;
        const float inv = (d > 0.0f) ? (1.0f / d) : 0.0f;   // nan_to_num(rel/dist)
        sUnit[tid][0] = rx * inv; sUnit[tid][1] = ry * inv; sUnit[tid][2] = rz * inv;
    }
    __syncthreads();

    // ---- Phase 2: build neighbor-specific pair features in LDS ----------
    for (int idx = tid; idx < M_ * KREST; idx += 256) {
        const int row = idx / KREST;
        const int col = idx - row * KREST;
        float v;
        if (col < D_) {
            v = emb[(size_t)sNid[row] * D_ + col];
        } else if (col < D_ + DD_) {
            const int k   = col - D_;
            const float f = __expf((float)(k & 15) * NEG_LOG_THETA_OVER_HM1);
            const float a = sDist[row] * f;
            v = (k < 16) ? __sinf(a) : __cosf(a);
        } else {
            v = edp[row * E_ + (col - (D_ + DD_))];
        }
        sXr[row][col] = v;
    }
    __syncthreads();

    // ---- Phase 3: GEMM1  H = relu(X @ we_w1 + b1),  X = [emb_i | sXr] ---
    {
        v8f c0 = {}; v8f c1 = {};
        // cols 0..127 of X are emb_i for every row -> broadcast A fragment
        for (int k0 = 0; k0 < D_; k0 += 4) {
            const int kk = k0 + 2 * half;
            v2f a;  a.x  = sEmb[kk];                    a.y  = sEmb[kk + 1];
            v2f bf; bf.x = we_w1[kk * MSG_ + col0 + nidx];
                    bf.y = we_w1[(kk + 1) * MSG_ + col0 + nidx];
            c0 = __builtin_amdgcn_wmma_f32_16x16x4_f32(false, a, false, bf, (short)0, c0, false, false);
            c1 = __builtin_amdgcn_wmma_f32_16x16x4_f32(false, a, false, bf, (short)0, c1, false, false);
        }
        const float* w1r = we_w1 + D_ * MSG_;           // rows 128..319
        for (int k0 = 0; k0 < KREST; k0 += 4) {
            const int kk = k0 + 2 * half;
            v2f a0; a0.x = sXr[nidx][kk];        a0.y = sXr[nidx][kk + 1];
            v2f a1; a1.x = sXr[16 + nidx][kk];   a1.y = sXr[16 + nidx][kk + 1];
            v2f bf; bf.x = w1r[kk * MSG_ + col0 + nidx];
                    bf.y = w1r[(kk + 1) * MSG_ + col0 + nidx];
            c0 = __builtin_amdgcn_wmma_f32_16x16x4_f32(false, a0, false, bf, (short)0, c0, false, false);
            c1 = __builtin_amdgcn_wmma_f32_16x16x4_f32(false, a1, false, bf, (short)0, c1, false, false);
        }
        const float bv = we_b1[col0 + nidx];
#pragma unroll
        for (int v = 0; v < 8; ++v) {
            const int r = v + 8 * half;
            sH[r][col0 + nidx]      = fmaxf(c0[v] + bv, 0.0f);
            sH[16 + r][col0 + nidx] = fmaxf(c1[v] + bv, 0.0f);
        }
    }
    __syncthreads();

    // ---- Phase 4: GEMM2  m_ij = H @ we_w2 + b2 --------------------------
    gemm32x128(&sH[0][0], we_w2, we_b2, &sM[0][0], false, half, nidx, col0);
    __syncthreads();

    // ---- Phase 5: GEMM3  Hx = relu(m_ij @ wx_w1 + bx1), reuse sXr -------
    float* sHx = &sXr[0][0];                            // 32x128 fits in sXr
    gemm32x128(&sM[0][0], wx_w1, wx_b1, sHx, true, half, nidx, col0);
    __syncthreads();

    // ---- Phase 6: per-neighbor scalar px, message mean ------------------
    if (tid < M_) {
        float acc = wx_b2[0];
        for (int k = 0; k < P_; ++k) acc += sHx[tid * P_ + k] * wx_w2[k];
        sPx[tid] = acc;
    }
    if (tid >= 32 && tid < 32 + MSG_) {
        const int dch = tid - 32;
        float s = 0.0f;
        for (int m = 0; m < M_; ++m) s += sM[m][dch];
        sMi[dch] = s * pmask[p] * (1.0f / (float)M_);
    }
    __syncthreads();

    // ---- Phase 7: coordinate update -------------------------------------
    if (tid < 3) {
        float acc = 0.0f;
        for (int m = 0; m < M_; ++m) acc += sUnit[m][tid] * sPx[m];
        out_coords[(size_t)p * 3 + tid] = cdp[tid] + acc * (1.0f / (float)M_);
    }

    // ---- Phase 8: node MLP (1x256 GEMV -> VALU) -------------------------
    if (tid < P_) {
        float acc = wh_b1[tid];
        for (int k = 0; k < D_;   ++k) acc += sEmb[k] * wh_w1[k * P_ + tid];
        for (int k = 0; k < MSG_; ++k) acc += sMi[k]  * wh_w1[(D_ + k) * P_ + tid];
        sT1[tid] = fmaxf(acc, 0.0f);
    }
    __syncthreads();
    if (tid < D_) {
        float acc = wh_b2[tid];
        for (int k = 0; k < P_; ++k) acc += sT1[k] * wh_w2[k * D_ + tid];
        out_emb[(size_t)p * D_ + tid] = sEmb[tid] + acc;
    }
}

extern "C" void kernel_launch(void* const* d_in, const int* in_sizes, int n_in,
                              void* d_out, int out_size, void* d_ws, size_t ws_size,
                              hipStream_t stream)
{
    (void)in_sizes; (void)n_in; (void)d_ws; (void)ws_size; (void)out_size;

    const float* emb    = (const float*)d_in[0];
    const float* coords = (const float*)d_in[1];
    const float* pmask  = (const float*)d_in[2];
    const float* edges  = (const float*)d_in[3];
    const float* we_w1  = (const float*)d_in[4];
    const float* we_b1  = (const float*)d_in[5];
    const float* we_w2  = (const float*)d_in[6];
    const float* we_b2  = (const float*)d_in[7];
    const float* wx_w1  = (const float*)d_in[8];
    const float* wx_b1  = (const float*)d_in[9];
    const float* wx_w2  = (const float*)d_in[10];
    const float* wx_b2  = (const float*)d_in[11];
    const float* wh_w1  = (const float*)d_in[12];
    const float* wh_b1  = (const float*)d_in[13];
    const float* wh_w2  = (const float*)d_in[14];
    const float* wh_b2  = (const float*)d_in[15];
    const int*   nbr    = (const int*)  d_in[16];

    float* out_emb    = (float*)d_out;                         // (B,N,D) flat
    float* out_coords = out_emb + (size_t)B_ * N_ * D_;        // (B,N,3) flat

    dim3 grid(B_ * N_);
    dim3 block(256);
    egnn_node_kernel<<<grid, block, 0, stream>>>(
        emb, coords, pmask, edges,
        we_w1, we_b1, we_w2, we_b2,
        wx_w1, wx_b1, wx_w2, wx_b2,
        wh_w1, wh_b1, wh_w2, wh_b2,
        nbr, out_emb, out_coords);
}